// S6_58660663328994
// MI455X (gfx1250) — compile-verified
//
#include <hip/hip_runtime.h>
#include <hip/hip_bf16.h>
#include <cstdint>

// ---------------------------------------------------------------------------
// Problem constants (match reference)
// ---------------------------------------------------------------------------
#define BB      2
#define LL      1024
#define DMODEL  2048
#define DINNER  4096
#define DSTATE  16
#define DCONV   4
#define DTRANK  128
#define MROWS   (BB * LL)          // 2048

typedef float v2f __attribute__((ext_vector_type(2)));
typedef float v8f __attribute__((ext_vector_type(8)));

// ---------------------------------------------------------------------------
// FP32 WMMA GEMM:  O[M x N] = A[M x K] * W[N x K]^T
//   - block: 128 threads = 4 wave32 waves; block tile 64(M) x 64(N)
//   - wave computes one 16-row strip x 64 cols (4 accumulators, A-frag reuse)
//   - K staged through LDS in 32-wide chunks, rows padded to 36 floats
//   - register double-buffer: next chunk's global loads are in flight while
//     the current chunk's 32 WMMAs issue from LDS
//   - M and K assumed multiples of 64 / 32 (true for all call sites);
//     N is bounds-guarded (needed for N=160)
// ---------------------------------------------------------------------------
#define GBM 64
#define GBN 64
#define GKC 32
#define LDSPAD 36

__global__ __launch_bounds__(128)
void gemm_wmma_f32(const float* __restrict__ A, int lda,
                   const float* __restrict__ W, int K,
                   float* __restrict__ O, int ldo,
                   int N)
{
    __shared__ float As[GBM][LDSPAD];
    __shared__ float Bs[GBN][LDSPAD];

    const int t    = threadIdx.x;        // 0..127
    const int wave = t >> 5;             // 0..3
    const int lane = t & 31;
    const int half = lane >> 4;          // 0: K lo pair, 1: K hi pair
    const int l16  = lane & 15;
    const int m0   = blockIdx.x * GBM;
    const int n0   = blockIdx.y * GBN;
    const int mloc = wave * 16;

    // per-thread fixed (row, col) slots for cooperative tile loads
    int rowi[4], c4i[4];
    bool wok[4];
    #pragma unroll
    for (int i = 0; i < 4; ++i) {
        int f4  = t + i * 128;           // 0..511 float4 slots
        rowi[i] = f4 >> 3;               // 8 float4 per 32-float row
        c4i[i]  = (f4 & 7) << 2;
        wok[i]  = (n0 + rowi[i]) < N;
    }

    v8f acc[4] = {};
    float4 pa[4], pb[4];

    auto load_chunk = [&](int kc) {
        #pragma unroll
        for (int i = 0; i < 4; ++i) {
            pa[i] = *(const float4*)(A + (size_t)(m0 + rowi[i]) * lda + kc + c4i[i]);
            pb[i] = wok[i] ? *(const float4*)(W + (size_t)(n0 + rowi[i]) * K + kc + c4i[i])
                           : make_float4(0.f, 0.f, 0.f, 0.f);
        }
    };
    auto store_chunk = [&]() {
        #pragma unroll
        for (int i = 0; i < 4; ++i) {
            *(float4*)&As[rowi[i]][c4i[i]] = pa[i];
            *(float4*)&Bs[rowi[i]][c4i[i]] = pb[i];
        }
    };

    load_chunk(0);

    for (int kc = 0; kc < K; kc += GKC) {
        store_chunk();
        __syncthreads();

        if (kc + GKC < K)
            load_chunk(kc + GKC);        // global loads overlap the WMMAs below

        #pragma unroll
        for (int kk = 0; kk < GKC; kk += 4) {
            v2f a = *(const v2f*)&As[mloc + l16][kk + 2 * half];
            #pragma unroll
            for (int nt = 0; nt < 4; ++nt) {
                v2f b = *(const v2f*)&Bs[nt * 16 + l16][kk + 2 * half];
                acc[nt] = __builtin_amdgcn_wmma_f32_16x16x4_f32(
                    /*neg_a=*/false, a, /*neg_b=*/false, b,
                    /*c_mod=*/(short)0, acc[nt],
                    /*reuse_a=*/false, /*reuse_b=*/false);
            }
        }
        __syncthreads();
    }

    // ---- store: VGPR v holds M = v (lanes 0-15) / 8+v (lanes 16-31) ----
    #pragma unroll
    for (int nt = 0; nt < 4; ++nt) {
        int col = n0 + nt * 16 + l16;
        if (col < N) {
            #pragma unroll
            for (int v = 0; v < 8; ++v) {
                int row = m0 + mloc + half * 8 + v;
                O[(size_t)row * ldo + col] = acc[nt][v];
            }
        }
    }
}

// ---------------------------------------------------------------------------
// Depthwise causal conv (k=4) + bias + SiLU.
// x is the low half of xz rows (stride 2*DINNER). One thread per (b,l,d).
// ---------------------------------------------------------------------------
__global__ __launch_bounds__(256)
void conv_silu_kernel(const float* __restrict__ xz,
                      const float* __restrict__ cw,
                      const float* __restrict__ cb,
                      float* __restrict__ xs)
{
    int gid = blockIdx.x * blockDim.x + threadIdx.x;     // < MROWS*DINNER
    int d   = gid & (DINNER - 1);
    int ml  = gid >> 12;                                 // b*L + l
    int l   = ml & (LL - 1);
    int b   = ml >> 10;

    float acc = cb[d];
    #pragma unroll
    for (int k = 0; k < DCONV; ++k) {
        int ls = l - (DCONV - 1) + k;
        if (ls >= 0)
            acc += xz[(size_t)(b * LL + ls) * (2 * DINNER) + d] * cw[d * DCONV + k];
    }
    // SiLU
    acc = acc / (1.f + expf(-acc));
    xs[gid] = acc;
}

// ---------------------------------------------------------------------------
// Selective scan. One thread per (b,d) channel; 16 states in registers;
// B(t)/C(t) broadcast through a double-buffered LDS staging area (one
// barrier per timestep). Fuses double-bias + softplus on dt (reference adds
// dt_proj_b twice) and the final y*z, writing y*z in place over x_silu.
// ---------------------------------------------------------------------------
__global__ __launch_bounds__(256)
void scan_kernel(const float* __restrict__ dt_raw,   // [M, DINNER] pre-bias
                 const float* __restrict__ dt_bias,  // [DINNER]
                 const float* __restrict__ xdbl,     // [M, 160] (B at 128, C at 144)
                 const float* __restrict__ xz,       // [M, 2*DINNER] (z = hi half)
                 const float* __restrict__ A_log,    // [DINNER, DSTATE]
                 const float* __restrict__ Dvec,     // [DINNER]
                 float* __restrict__ xsy)            // in: silu(x); out: y*z
{
    __shared__ float sBC[2][2 * DSTATE];             // double-buffered B|C

    const int b = blockIdx.x >> 4;                   // 4096/256 = 16 blocks per batch
    const int d = (blockIdx.x & 15) * 256 + threadIdx.x;

    float a[DSTATE], h[DSTATE];
    #pragma unroll
    for (int n = 0; n < DSTATE; ++n) {
        a[n] = -expf(A_log[d * DSTATE + n]);
        h[n] = 0.f;
    }
    const float bias2 = 2.f * dt_bias[d];
    const float Dd    = Dvec[d];

    for (int l = 0; l < LL; ++l) {
        const int ml  = b * LL + l;
        const int buf = l & 1;
        if (threadIdx.x < 2 * DSTATE)
            sBC[buf][threadIdx.x] =
                xdbl[(size_t)ml * (DTRANK + 2 * DSTATE) + DTRANK + threadIdx.x];
        __syncthreads();   // parity buffer: safe with a single barrier per step

        float dtv = dt_raw[(size_t)ml * DINNER + d] + bias2;
        dtv = (dtv > 20.f) ? dtv : log1pf(expf(dtv));          // softplus
        const float xv = xsy[(size_t)ml * DINNER + d];

        float y = 0.f;
        #pragma unroll
        for (int n = 0; n < DSTATE; ++n) {
            float dA = expf(dtv * a[n]);
            h[n] = dA * h[n] + (dtv * sBC[buf][n]) * xv;
            y   += h[n] * sBC[buf][DSTATE + n];
        }
        y += xv * Dd;
        const float zv = xz[(size_t)ml * (2 * DINNER) + DINNER + d];
        xsy[(size_t)ml * DINNER + d] = y * zv;
    }
}

// ---------------------------------------------------------------------------
// Launch
// ---------------------------------------------------------------------------
extern "C" void kernel_launch(void* const* d_in, const int* in_sizes, int n_in,
                              void* d_out, int out_size, void* d_ws, size_t ws_size,
                              hipStream_t stream)
{
    const float* hidden   = (const float*)d_in[0];
    const float* in_proj  = (const float*)d_in[1];
    const float* conv_w   = (const float*)d_in[2];
    const float* conv_b   = (const float*)d_in[3];
    const float* x_proj   = (const float*)d_in[4];
    const float* dt_proj  = (const float*)d_in[5];
    const float* dt_bias  = (const float*)d_in[6];
    const float* A_log    = (const float*)d_in[7];
    const float* Dvec     = (const float*)d_in[8];
    const float* out_proj = (const float*)d_in[9];
    float*       out      = (float*)d_out;

    float* ws   = (float*)d_ws;
    float* xz   = ws;                                        // [M, 8192]
    float* xsy  = xz  + (size_t)MROWS * (2 * DINNER);        // [M, 4096]
    float* xdbl = xsy + (size_t)MROWS * DINNER;              // [M, 160]
    float* dtr  = xdbl + (size_t)MROWS * (DTRANK + 2 * DSTATE); // [M, 4096]

    // 1) xz = H * in_proj^T          M=2048 K=2048 N=8192
    gemm_wmma_f32<<<dim3(MROWS / GBM, (2 * DINNER) / GBN), 128, 0, stream>>>(
        hidden, DMODEL, in_proj, DMODEL, xz, 2 * DINNER, 2 * DINNER);

    // 2) depthwise conv + bias + SiLU
    conv_silu_kernel<<<(MROWS * DINNER) / 256, 256, 0, stream>>>(
        xz, conv_w, conv_b, xsy);

    // 3) x_dbl = x_silu * x_proj^T   M=2048 K=4096 N=160
    gemm_wmma_f32<<<dim3(MROWS / GBM, (160 + GBN - 1) / GBN), 128, 0, stream>>>(
        xsy, DINNER, x_proj, DINNER, xdbl, DTRANK + 2 * DSTATE, DTRANK + 2 * DSTATE);

    // 4) dt_raw = x_dbl[:, :128] * dt_proj^T   M=2048 K=128 N=4096 (lda=160)
    gemm_wmma_f32<<<dim3(MROWS / GBM, DINNER / GBN), 128, 0, stream>>>(
        xdbl, DTRANK + 2 * DSTATE, dt_proj, DTRANK, dtr, DINNER, DINNER);

    // 5) selective scan (fused softplus + y*z, in place over xsy)
    scan_kernel<<<BB * (DINNER / 256), 256, 0, stream>>>(
        dtr, dt_bias, xdbl, xz, A_log, Dvec, xsy);

    // 6) out = y * out_proj^T        M=2048 K=4096 N=2048
    gemm_wmma_f32<<<dim3(MROWS / GBM, DMODEL / GBN), 128, 0, stream>>>(
        xsy, DINNER, out_proj, DINNER, out, DMODEL, DMODEL);
}